// DynamicPool3d_63316407877670
// MI455X (gfx1250) — compile-verified
//
#include <hip/hip_runtime.h>
#include <stdint.h>

// Input  x  : (1, 64, 128, 128, 128) fp32
// Output out: (1, 64,  64,  64,  64) fp32
#define CSTR  2097152   // 128^3  input channel stride
#define DSTR  16384     // 128^2  input depth stride
#define HSTR  128       //        input row stride
#define OCS   262144    // 64^3   output channel stride
#define ODS   4096      // 64^2   output depth stride
#define OHS   64        //        output row stride

__device__ __forceinline__ void async_ld_b128(uint32_t lds_byte_off, const void* gptr) {
    // VDST = LDS byte address VGPR, VADDR = 64-bit global address VGPR pair (GV mode).
    asm volatile("global_load_async_to_lds_b128 %0, %1, off"
                 :: "v"(lds_byte_off), "v"(gptr) : "memory");
}
__device__ __forceinline__ void wait_async_all() {
    asm volatile("s_wait_asynccnt 0" ::: "memory");
}

__global__ __launch_bounds__(256) void dynpool3d_kernel(const float* __restrict__ x,
                                                        float* __restrict__ out) {
    __shared__ float smem[12288];                       // 48 KB max (part B)
    const uint32_t lds_base = (uint32_t)(uintptr_t)(&smem[0]);
    const int tid = threadIdx.x;
    const int b   = blockIdx.x;

    if (b < 8192) {
        // ---------- Part A: channels 0..15, plain 2x2x2 max pool ----------
        const int c   = b >> 9;           // 0..15
        const int rem = b & 511;
        const int d0  = rem >> 3;         // output depth slice 0..63
        const int h0  = (rem & 7) << 3;   // 8 output rows per block
        const float* base = x + (size_t)c * CSTR + (size_t)(2 * d0) * DSTR
                              + (size_t)(2 * h0) * HSTR;
        // Stage 2 d-slices x 16 h-rows x 128 w = 16 KB as 1024 b128 chunks.
        #pragma unroll
        for (int r = 0; r < 4; ++r) {
            int i   = tid + (r << 8);     // 0..1023
            int row = i >> 5;             // dd*16 + hh  (0..31)
            int w4  = i & 31;             // 16B chunk within row
            const float* g = base + (size_t)(row >> 4) * DSTR
                                  + (size_t)(row & 15) * HSTR + (w4 << 2);
            async_ld_b128(lds_base + (uint32_t)(i << 4), g);
        }
        wait_async_all();
        __syncthreads();

        const float2* s2 = (const float2*)smem;         // [dd][hh][64 x float2]
        float* o = out + (size_t)c * OCS + (size_t)d0 * ODS + (size_t)h0 * OHS;
        #pragma unroll
        for (int pp = 0; pp < 2; ++pp) {
            int p  = tid + (pp << 8);
            int ho = p >> 6, wo = p & 63;
            float m = -__builtin_inff();
            #pragma unroll
            for (int kd = 0; kd < 2; ++kd)
                #pragma unroll
                for (int kh = 0; kh < 2; ++kh) {
                    float2 v = s2[(kd * 16 + 2 * ho + kh) * 64 + wo];
                    m = fmaxf(m, fmaxf(v.x, v.y));
                }
            __builtin_nontemporal_store(m, &o[ho * OHS + wo]);
        }
    } else {
        // ---------- Part B: 16 groups of 3 channels, argmax of sum-of-squares ----------
        const int bb  = b - 8192;
        const int grp = bb >> 9;          // 0..15
        const int rem = bb & 511;
        const int d0  = rem >> 3;
        const int h0  = (rem & 7) << 3;
        const float* base = x + (size_t)(16 + 3 * grp) * CSTR
                              + (size_t)(2 * d0) * DSTR + (size_t)(2 * h0) * HSTR;
        // Stage 3 channels x 2 d-slices x 16 h-rows x 128 w = 48 KB as 3072 b128 chunks.
        #pragma unroll
        for (int r = 0; r < 12; ++r) {
            int i   = tid + (r << 8);     // 0..3071
            int row = i >> 5;             // c*32 + dd*16 + hh  (0..95)
            int w4  = i & 31;
            int cc  = row >> 5;
            int rr  = row & 31;
            const float* g = base + (size_t)cc * CSTR + (size_t)(rr >> 4) * DSTR
                                  + (size_t)(rr & 15) * HSTR + (w4 << 2);
            async_ld_b128(lds_base + (uint32_t)(i << 4), g);
        }
        wait_async_all();
        __syncthreads();

        const float2* s2 = (const float2*)smem;         // [c][dd][hh][64 x float2]
        float* o = out + (size_t)(16 + 3 * grp) * OCS + (size_t)d0 * ODS
                       + (size_t)h0 * OHS;
        #pragma unroll
        for (int pp = 0; pp < 2; ++pp) {
            int p  = tid + (pp << 8);
            int ho = p >> 6, wo = p & 63;
            float nrm[8];
            #pragma unroll
            for (int j = 0; j < 8; ++j) nrm[j] = 0.0f;
            #pragma unroll
            for (int cc = 0; cc < 3; ++cc)
                #pragma unroll
                for (int kd = 0; kd < 2; ++kd)
                    #pragma unroll
                    for (int kh = 0; kh < 2; ++kh) {
                        float2 v = s2[(cc * 32 + kd * 16 + 2 * ho + kh) * 64 + wo];
                        int j0 = kd * 4 + kh * 2;
                        nrm[j0]     = fmaf(v.x, v.x, nrm[j0]);
                        nrm[j0 + 1] = fmaf(v.y, v.y, nrm[j0 + 1]);
                    }
            // First-occurrence argmax (matches jnp.argmax): strict '>' keeps earliest max.
            int jb = 0; float best = nrm[0];
            #pragma unroll
            for (int j = 1; j < 8; ++j) {
                if (nrm[j] > best) { best = nrm[j]; jb = j; }
            }
            const int kd = jb >> 2, kh = (jb >> 1) & 1, kw = jb & 1;
            #pragma unroll
            for (int cc = 0; cc < 3; ++cc) {
                float v = smem[(cc * 32 + kd * 16 + 2 * ho + kh) * 128 + 2 * wo + kw];
                __builtin_nontemporal_store(v, &o[(size_t)cc * OCS + ho * OHS + wo]);
            }
        }
    }
}

extern "C" void kernel_launch(void* const* d_in, const int* in_sizes, int n_in,
                              void* d_out, int out_size, void* d_ws, size_t ws_size,
                              hipStream_t stream) {
    const float* x  = (const float*)d_in[0];
    float* out      = (float*)d_out;
    (void)in_sizes; (void)n_in; (void)d_ws; (void)ws_size; (void)out_size;
    // 8192 maxpool blocks + 8192 group-argmax blocks, 256 threads each.
    hipLaunchKernelGGL(dynpool3d_kernel, dim3(16384), dim3(256), 0, stream, x, out);
}